// DynamicGNN_88948772700690
// MI455X (gfx1250) — compile-verified
//
#include <hip/hip_runtime.h>
#include <hip/hip_bf16.h>

typedef __attribute__((ext_vector_type(2))) float v2f;
typedef __attribute__((ext_vector_type(8))) float v8f;

#define NN 50000
#define EE 600000
#define FF 128

// workspace layout, in floats
#define WS_WN   0                      // 128*128 evolved GCN weight
#define WS_DEG  16384                  // NN degrees -> overwritten with dinv
#define WS_XT   66560                  // NN*FF  (x @ W.T)
#define WS_AGG  (66560 + NN * FF)      // NN*FF  scatter accumulator

// ---------------------------------------------------------------------------
// 0) init: agg = 0, deg = 1.0 (self-loop weight pre-added)
// ---------------------------------------------------------------------------
__global__ void k_init(float* __restrict__ ws) {
    int t = blockIdx.x * blockDim.x + threadIdx.x;
    if (t < NN * FF) ws[WS_AGG + t] = 0.0f;
    if (t < NN)      ws[WS_DEG + t] = 1.0f;
}

// ---------------------------------------------------------------------------
// 1) GRU step on the 128x128 weight (tiny: 16384 threads x 768 MACs)
// ---------------------------------------------------------------------------
__global__ void k_gru(const float* __restrict__ W,
                      const float* __restrict__ wih, const float* __restrict__ whh,
                      const float* __restrict__ bih, const float* __restrict__ bhh,
                      float* __restrict__ Wn) {
    int idx = blockIdx.x * blockDim.x + threadIdx.x;
    if (idx >= FF * FF) return;
    int i = idx >> 7, j = idx & 127;
    const float* wr = W + i * FF;
    float gir = bih[j], giz = bih[j + FF], gin = bih[j + 2 * FF];
    float ghr = bhh[j], ghz = bhh[j + FF], ghn = bhh[j + 2 * FF];
    const float* a0 = wih + j * FF;
    const float* a1 = wih + (j + FF) * FF;
    const float* a2 = wih + (j + 2 * FF) * FF;
    const float* h0 = whh + j * FF;
    const float* h1 = whh + (j + FF) * FF;
    const float* h2 = whh + (j + 2 * FF) * FF;
#pragma unroll 4
    for (int k = 0; k < FF; ++k) {
        float wk = wr[k];
        gir = fmaf(wk, a0[k], gir);
        giz = fmaf(wk, a1[k], giz);
        gin = fmaf(wk, a2[k], gin);
        ghr = fmaf(wk, h0[k], ghr);
        ghz = fmaf(wk, h1[k], ghz);
        ghn = fmaf(wk, h2[k], ghn);
    }
    float r  = 1.0f / (1.0f + __expf(-(gir + ghr)));
    float z  = 1.0f / (1.0f + __expf(-(giz + ghz)));
    float nv = tanhf(gin + r * ghn);
    Wn[idx] = (1.0f - z) * nv + z * wr[j];
}

// ---------------------------------------------------------------------------
// 2) degree accumulation over real edges (deg pre-initialized to 1.0)
// ---------------------------------------------------------------------------
__global__ void k_deg(const int* __restrict__ ei, const float* __restrict__ ew,
                      float* __restrict__ deg) {
    int e = blockIdx.x * blockDim.x + threadIdx.x;
    if (e >= EE) return;
    atomicAdd(&deg[ei[EE + e]], ew[e]);   // dst = edge_index[1]
}

// ---------------------------------------------------------------------------
// 3) dinv = deg > 0 ? rsqrt(deg) : 0   (in place)
// ---------------------------------------------------------------------------
__global__ void k_dinv(float* __restrict__ deg) {
    int n = blockIdx.x * blockDim.x + threadIdx.x;
    if (n >= NN) return;
    float d = deg[n];
    deg[n] = d > 0.0f ? __frsqrt_rn(d) : 0.0f;
}

// ---------------------------------------------------------------------------
// 4) xt = x @ Wn.T via V_WMMA_F32_16X16X4_F32.
//    Block = 8 waves; each wave computes a 16(M) x 128(N) strip.
//    Wn (64KB) staged in LDS once per block; B-fragments come from LDS.
//    A layout (16x4 f32): lane holds A[lane%16][2*(lane/16) + {0,1}]
//    B layout (4x16 f32): lane holds B[2*(lane/16) + {0,1}][lane%16]
//    D layout: VGPR v -> row (v + 8*(lane/16)), col lane%16
// ---------------------------------------------------------------------------
__global__ void __launch_bounds__(256)
k_xt_wmma(const float* __restrict__ x, const float* __restrict__ Wn,
          float* __restrict__ xt) {
    __shared__ float Wlds[FF * FF];   // 64 KB (WGP has 320 KB)

    // cooperative stage of Wn into LDS (float4)
    {
        const float4* src = (const float4*)Wn;
        float4* dst = (float4*)Wlds;
#pragma unroll
        for (int i = 0; i < (FF * FF / 4) / 256; ++i)
            dst[threadIdx.x + i * 256] = src[threadIdx.x + i * 256];
    }
    __syncthreads();

    const int wave = threadIdx.x >> 5;
    const int lane = threadIdx.x & 31;
    const int tile = blockIdx.x * 8 + wave;          // 16-row tile of x
    if (tile * 16 >= NN) return;                      // NN % 16 == 0: whole waves only

    const int mcol = lane & 15;       // M for A, N for B/D
    const int half = lane >> 4;       // lane group

    const float* xrow = x + (size_t)(tile * 16 + mcol) * FF + 2 * half;
    const float* brow = Wlds + (size_t)mcol * FF + 2 * half;   // + nt*16*FF per N-tile

    v8f acc[8];
#pragma unroll
    for (int nt = 0; nt < 8; ++nt) acc[nt] = (v8f){0,0,0,0,0,0,0,0};

#pragma unroll
    for (int kk = 0; kk < 32; ++kk) {                 // K = 128 in steps of 4
        v2f a = *(const v2f*)(xrow + 4 * kk);
#pragma unroll
        for (int nt = 0; nt < 8; ++nt) {
            v2f b = *(const v2f*)(brow + (size_t)nt * 16 * FF + 4 * kk);
            acc[nt] = __builtin_amdgcn_wmma_f32_16x16x4_f32(
                false, a, false, b, (short)0, acc[nt], false, false);
        }
    }

    // scatter D to xt: element v of acc[nt] -> row (v + 8*half), col (nt*16 + mcol)
#pragma unroll
    for (int nt = 0; nt < 8; ++nt) {
#pragma unroll
        for (int v = 0; v < 8; ++v) {
            xt[(size_t)(tile * 16 + v + 8 * half) * FF + nt * 16 + mcol] = acc[nt][v];
        }
    }
}

// ---------------------------------------------------------------------------
// 5) edge scatter: one wave per edge, one float4 of features per lane
// ---------------------------------------------------------------------------
__global__ void __launch_bounds__(256)
k_scatter(const int* __restrict__ ei, const float* __restrict__ ew,
          const float* __restrict__ dinv, const float* __restrict__ xt,
          float* __restrict__ agg) {
    int t = blockIdx.x * blockDim.x + threadIdx.x;
    int e = t >> 5;
    int lane = t & 31;
    if (e >= EE) return;
    int s = ei[e];
    int d = ei[EE + e];
    float w = dinv[s] * ew[e] * dinv[d];
    float4 v = ((const float4*)(xt + (size_t)s * FF))[lane];
    float* ap = agg + (size_t)d * FF + lane * 4;
    atomicAdd(ap + 0, v.x * w);
    atomicAdd(ap + 1, v.y * w);
    atomicAdd(ap + 2, v.z * w);
    atomicAdd(ap + 3, v.w * w);
}

// ---------------------------------------------------------------------------
// 6) finalize: add self-loop term, ReLU, dot with w_lin, + b_lin
//    one wave per node, shuffle reduction
// ---------------------------------------------------------------------------
__global__ void __launch_bounds__(256)
k_final(const float* __restrict__ agg, const float* __restrict__ xt,
        const float* __restrict__ dinv, const float* __restrict__ w_lin,
        const float* __restrict__ b_lin, float* __restrict__ out) {
    int t = blockIdx.x * blockDim.x + threadIdx.x;
    int n = t >> 5;
    int lane = t & 31;
    if (n >= NN) return;
    float di = dinv[n];
    float sw = di * di;                                // self-loop norm
    float4 a  = ((const float4*)(agg + (size_t)n * FF))[lane];
    float4 xv = ((const float4*)(xt  + (size_t)n * FF))[lane];
    float4 wl = ((const float4*)w_lin)[lane];
    float4 h;
    h.x = fmaxf(a.x + xv.x * sw, 0.0f);
    h.y = fmaxf(a.y + xv.y * sw, 0.0f);
    h.z = fmaxf(a.z + xv.z * sw, 0.0f);
    h.w = fmaxf(a.w + xv.w * sw, 0.0f);
    float p = h.x * wl.x + h.y * wl.y + h.z * wl.z + h.w * wl.w;
#pragma unroll
    for (int off = 16; off > 0; off >>= 1)
        p += __shfl_down(p, off, 32);
    if (lane == 0) out[n] = p + b_lin[0];
}

// ---------------------------------------------------------------------------
extern "C" void kernel_launch(void* const* d_in, const int* in_sizes, int n_in,
                              void* d_out, int out_size, void* d_ws, size_t ws_size,
                              hipStream_t stream) {
    const float* x     = (const float*)d_in[0];
    const int*   ei    = (const int*)  d_in[1];
    const float* ew    = (const float*)d_in[2];
    const float* Wg    = (const float*)d_in[3];
    const float* wih   = (const float*)d_in[4];
    const float* whh   = (const float*)d_in[5];
    const float* bih   = (const float*)d_in[6];
    const float* bhh   = (const float*)d_in[7];
    const float* w_lin = (const float*)d_in[8];
    const float* b_lin = (const float*)d_in[9];
    float* out = (float*)d_out;
    float* ws  = (float*)d_ws;

    float* Wn   = ws + WS_WN;
    float* dinv = ws + WS_DEG;
    float* xt   = ws + WS_XT;
    float* agg  = ws + WS_AGG;

    // 0) init accumulator + degree
    k_init<<<(NN * FF + 255) / 256, 256, 0, stream>>>(ws);
    // 1) GRU evolve weight
    k_gru<<<(FF * FF + 255) / 256, 256, 0, stream>>>(Wg, wih, whh, bih, bhh, Wn);
    // 2) degree accumulation
    k_deg<<<(EE + 255) / 256, 256, 0, stream>>>(ei, ew, dinv);
    // 3) dinv
    k_dinv<<<(NN + 255) / 256, 256, 0, stream>>>(dinv);
    // 4) xt = x @ Wn.T  (WMMA)
    {
        int tiles = NN / 16;                   // 3125
        int blocks = (tiles + 7) / 8;          // 391
        k_xt_wmma<<<blocks, 256, 0, stream>>>(x, Wn, xt);
    }
    // 5) edge scatter
    k_scatter<<<(EE * 32 + 255) / 256, 256, 0, stream>>>(ei, ew, dinv, xt, agg);
    // 6) head
    k_final<<<(NN * 32 + 255) / 256, 256, 0, stream>>>(agg, xt, dinv, w_lin, b_lin, out);
}